// AdaClusteringAttention_82867099009675
// MI455X (gfx1250) — compile-verified
//
#include <hip/hip_runtime.h>
#include <hip/hip_bf16.h>

typedef __attribute__((ext_vector_type(2))) float v2f;
typedef __attribute__((ext_vector_type(8))) float v8f;

#define NB   32          // batch (heads)
#define BC   4           // cluster batch
#define NN   4096        // sequence length
#define DD   64          // head dim
#define CC   129         // clusters + 1
#define CPAD 144         // 9 tiles of 16
#define CSTRIDE (CC * DD)  // 8256 floats per [129][64] center matrix

// ---------------------------------------------------------------------------
// Kernel 1: per-cluster-batch histogram -> float counts [4][129]
// ---------------------------------------------------------------------------
__global__ void counts_kernel(const int* __restrict__ clusters,
                              float* __restrict__ counts) {
    __shared__ int hist[CC];
    const int bc = blockIdx.x;
    const int tid = threadIdx.x;
    for (int i = tid; i < CC; i += 256) hist[i] = 0;
    __syncthreads();
    const int* g = clusters + bc * NN;
    for (int n = tid; n < NN; n += 256) atomicAdd(&hist[g[n]], 1);
    __syncthreads();
    for (int i = tid; i < CC; i += 256) counts[bc * CC + i] = (float)hist[i];
}

// ---------------------------------------------------------------------------
// Kernel 2: segment-mean into centers. grid = (32 batches, 3 matrices).
// centers layout: [3][32][129*64] floats
// ---------------------------------------------------------------------------
__global__ void segsum_kernel(const float* __restrict__ q,
                              const float* __restrict__ k,
                              const float* __restrict__ v,
                              const int* __restrict__ clusters,
                              const float* __restrict__ counts,
                              float* __restrict__ centers) {
    __shared__ float acc[CSTRIDE];  // 33 KB
    const int b = blockIdx.x;
    const int which = blockIdx.y;
    const int tid = threadIdx.x;
    const float* src = (which == 0 ? q : which == 1 ? k : v) + (size_t)b * NN * DD;
    const int* g = clusters + (b & (BC - 1)) * NN;

    for (int i = tid; i < CSTRIDE; i += 256) acc[i] = 0.f;
    __syncthreads();
    for (int i = tid; i < NN * DD; i += 256) {
        const int n = i >> 6;
        const int d = i & 63;
        atomicAdd(&acc[g[n] * DD + d], src[i]);   // ds_add_f32
    }
    __syncthreads();
    float* dst = centers + ((size_t)which * NB + b) * CSTRIDE;
    const float* cnt = counts + (b & (BC - 1)) * CC;
    for (int i = tid; i < CSTRIDE; i += 256) {
        const float c = cnt[i >> 6];
        dst[i] = (c > 0.f) ? acc[i] / c : 0.f;
    }
}

// ---------------------------------------------------------------------------
// Kernel 3: per-batch center attention via V_WMMA_F32_16X16X4_F32 + gather.
// One workgroup (8 wave32) per batch b.
// ---------------------------------------------------------------------------
__global__ void attn_kernel(const float* __restrict__ cent_q,
                            const float* __restrict__ cent_k,
                            const float* __restrict__ cent_v,
                            const float* __restrict__ counts,
                            const int* __restrict__ clusters,
                            float* __restrict__ out,
                            float* __restrict__ out2) {
    __shared__ float Qs[CPAD * DD];      // 36 KB, reused for Vc
    __shared__ float Ks[CPAD * DD];      // 36 KB, reused for O
    __shared__ float Ss[CPAD * CPAD];    // 81 KB, S -> A_full

    const int b = blockIdx.x;
    const int bc = b & (BC - 1);
    const int tid = threadIdx.x;
    const int lane = tid & 31;
    const int wave = tid >> 5;
    const int l16 = lane & 15;
    const int koff = (lane >> 4) << 1;   // 0 for lanes 0-15, 2 for lanes 16-31

    // ---- Phase A: stage Qc, Kc (zero-pad rows 129..143) -------------------
    const float* Qc = cent_q + (size_t)b * CSTRIDE;
    const float* Kc = cent_k + (size_t)b * CSTRIDE;
    for (int i = tid; i < CPAD * DD; i += 256) {
        const int r = i >> 6;
        Qs[i] = (r < CC) ? Qc[i] : 0.f;
        Ks[i] = (r < CC) ? Kc[i] : 0.f;
    }
    __syncthreads();

    // ---- Phase B: S = Qc * Kc^T  (9x9 tiles of 16x16, K = 64) -------------
    for (int t = wave; t < 81; t += 8) {
        const int mt = t / 9, nt = t % 9;
        v8f acc = {};
        const float* qrow = &Qs[(mt * 16 + l16) * DD];
        const float* krow = &Ks[(nt * 16 + l16) * DD];
        for (int kb = 0; kb < DD; kb += 4) {
            v2f a, bb;
            a.x  = qrow[kb + koff];     a.y  = qrow[kb + koff + 1];
            bb.x = krow[kb + koff];     bb.y = krow[kb + koff + 1];
            acc = __builtin_amdgcn_wmma_f32_16x16x4_f32(
                false, a, false, bb, (short)0, acc, false, false);
        }
        const int rbase = mt * 16 + ((lane >> 4) << 3);
        const int col = nt * 16 + l16;
#pragma unroll
        for (int vv = 0; vv < 8; ++vv) Ss[(rbase + vv) * CPAD + col] = acc[vv];
    }
    __syncthreads();

    // ---- Phase C: softmax with count reweighting (thread-per-row) ---------
    const float* cnt = counts + bc * CC;
    for (int r = tid; r < CC; r += 256) {
        float* srow = &Ss[r * CPAD];
        float mx = -3.402823466e38f;
        for (int c = 0; c < CC; ++c) mx = fmaxf(mx, srow[c]);
        float sum = 0.f;
        for (int c = 0; c < CC; ++c) {
            const float e = __expf(srow[c] - mx) * cnt[c];
            srow[c] = e;
            sum += e;
        }
        const float inv = 1.f / sum;           // sum > 0 (counts sum to N)
        for (int c = 0; c < CC; ++c) srow[c] *= inv;
        out2[b * CC + r] = srow[0];            // A_full[:, :, 0]
    }
    __syncthreads();

    // ---- Phase D: load Vc into Qs (zero-padded) ---------------------------
    const float* Vc = cent_v + (size_t)b * CSTRIDE;
    for (int i = tid; i < CPAD * DD; i += 256) {
        const int r = i >> 6;
        Qs[i] = (r < CC) ? Vc[i] : 0.f;
    }
    __syncthreads();

    // ---- Phase E: O = A_full * Vc (9x4 tiles, K = 144, pads are zero) -----
    for (int t = wave; t < 36; t += 8) {
        const int mt = t >> 2, nt = t & 3;
        v8f acc = {};
        const float* arow = &Ss[(mt * 16 + l16) * CPAD];
        for (int kb = 0; kb < CPAD; kb += 4) {
            v2f a, bb;
            a.x  = arow[kb + koff];                       a.y  = arow[kb + koff + 1];
            bb.x = Qs[(kb + koff) * DD + nt * 16 + l16];  bb.y = Qs[(kb + koff + 1) * DD + nt * 16 + l16];
            acc = __builtin_amdgcn_wmma_f32_16x16x4_f32(
                false, a, false, bb, (short)0, acc, false, false);
        }
        const int rbase = mt * 16 + ((lane >> 4) << 3);
        const int col = nt * 16 + l16;
#pragma unroll
        for (int vv = 0; vv < 8; ++vv) Ks[(rbase + vv) * DD + col] = acc[vv];
    }
    __syncthreads();

    // ---- Phase F: gather centers back to all tokens (coalesced store) -----
    const int* g = clusters + bc * NN;
    float* ob = out + (size_t)b * NN * DD;
    for (int i = tid; i < NN * DD; i += 256) {
        const int n = i >> 6;
        const int d = i & 63;
        ob[i] = Ks[g[n] * DD + d];
    }
}

// ---------------------------------------------------------------------------
extern "C" void kernel_launch(void* const* d_in, const int* in_sizes, int n_in,
                              void* d_out, int out_size, void* d_ws, size_t ws_size,
                              hipStream_t stream) {
    const float* q = (const float*)d_in[0];
    const float* k = (const float*)d_in[1];
    const float* v = (const float*)d_in[2];
    const int* clusters = (const int*)d_in[3];

    float* out  = (float*)d_out;
    float* out2 = out + (size_t)NB * NN * DD;   // second output: A_full[:, :, 0]

    // ws layout (floats): [0,1024)   counts [4][129]
    //                     [1024,...) centers [3][32][129*64]
    float* counts  = (float*)d_ws;
    float* centers = counts + 1024;

    counts_kernel<<<4, 256, 0, stream>>>(clusters, counts);
    segsum_kernel<<<dim3(NB, 3), 256, 0, stream>>>(q, k, v, clusters, counts, centers);
    attn_kernel<<<NB, 256, 0, stream>>>(centers,
                                        centers + (size_t)NB * CSTRIDE,
                                        centers + (size_t)2 * NB * CSTRIDE,
                                        counts, clusters, out, out2);
}